// Encoder_24438363914778
// MI455X (gfx1250) — compile-verified
//
#include <hip/hip_runtime.h>

// ---------------------------------------------------------------------------
// GraphSAGE VAE encoder for MI455X (gfx1250, wave32, WMMA).
// GEMMs: V_WMMA_F32_16X16X4_F32, double-buffered LDS staging via
// GLOBAL_LOAD_ASYNC_TO_LDS_B128 (ASYNCcnt) when the toolchain exposes it.
// ---------------------------------------------------------------------------

typedef float v2f __attribute__((ext_vector_type(2)));
typedef float v8f __attribute__((ext_vector_type(8)));
typedef int b128i __attribute__((vector_size(16)));  // builtin's 128-bit unit

#define N_NODES 10000
#define N_EDGES 80000
#define KB 16  // K-step per LDS stage (all layer dims are multiples of 16)

#if defined(__AMDGCN__) && __has_builtin(__builtin_amdgcn_global_load_async_to_lds_b128)
#define USE_ASYNC_LDS 1
#else
#define USE_ASYNC_LDS 0
#endif

#if USE_ASYNC_LDS
#define GPTR(p) ((b128i*)(p))
#define LPTR(p) ((b128i*)(p))
__device__ __forceinline__ void async_wait_le2() {
#if __has_builtin(__builtin_amdgcn_s_wait_asynccnt)
  __builtin_amdgcn_s_wait_asynccnt(2);
#else
  asm volatile("s_wait_asynccnt 0x2" ::: "memory");
#endif
}
__device__ __forceinline__ void async_wait_le0() {
#if __has_builtin(__builtin_amdgcn_s_wait_asynccnt)
  __builtin_amdgcn_s_wait_asynccnt(0);
#else
  asm volatile("s_wait_asynccnt 0x0" ::: "memory");
#endif
}
#endif

// ---------------- degree / reciprocal-count ----------------
__global__ void count_edges_kernel(const int* __restrict__ dst,
                                   float* __restrict__ cnt) {
  int e = blockIdx.x * blockDim.x + threadIdx.x;
  if (e < N_EDGES) atomicAdd(&cnt[dst[e]], 1.0f);
}

__global__ void make_rcnt_kernel(float* __restrict__ cnt) {
  int n = blockIdx.x * blockDim.x + threadIdx.x;
  if (n < N_NODES) cnt[n] = 1.0f / fmaxf(cnt[n], 1.0f);
}

// ---------------- mean aggregation: scatter-add ----------------
__global__ void scatter_add_kernel(const float* __restrict__ h,
                                   const int* __restrict__ src,
                                   const int* __restrict__ dst,
                                   float* __restrict__ agg, int din4) {
  long long tid = (long long)blockIdx.x * blockDim.x + threadIdx.x;
  long long total = (long long)N_EDGES * din4;
  if (tid >= total) return;
  int e = (int)(tid / din4);
  int f = (int)(tid % din4) * 4;
  int din = din4 * 4;
  const float4 v = *(const float4*)(h + (size_t)src[e] * din + f);
  float* a = agg + (size_t)dst[e] * din + f;
  atomicAdd(a + 0, v.x);
  atomicAdd(a + 1, v.y);
  atomicAdd(a + 2, v.z);
  atomicAdd(a + 3, v.w);
}

__global__ void scale_rows_kernel(float* __restrict__ agg,
                                  const float* __restrict__ rcnt, int din) {
  long long tid = (long long)blockIdx.x * blockDim.x + threadIdx.x;
  int din4 = din >> 2;
  long long total = (long long)N_NODES * din4;
  if (tid >= total) return;
  int n = (int)(tid / din4);
  int f = (int)(tid % din4) * 4;
  float s = rcnt[n];
  float4 v = *(float4*)(agg + (size_t)n * din + f);
  v.x *= s; v.y *= s; v.z *= s; v.w *= s;
  *(float4*)(agg + (size_t)n * din + f) = v;
}

// ---------------- WMMA GEMM: out = agg@Wl + h@Wr + b (+ReLU) ----------------
// Block = 256 threads = 8 waves arranged 4(M) x 2(N): block tile 64x64.
// Each wave owns a 16x32 output tile = two 16x16 fp32 accumulators that share
// one A fragment (3 LDS dwords per WMMA). K-loop runs twice
// (phase 0: A=agg,W=Wl; phase 1: A=h,W=Wr) accumulating into the same tiles.
// Tiles double-buffer through LDS; with async-to-LDS the next stage is kept in
// flight behind the WMMAs (s_wait_asynccnt <= 2).
__global__ __launch_bounds__(256) void sage_gemm_kernel(
    const float* __restrict__ Agg, const float* __restrict__ H,
    const float* __restrict__ Wl, const float* __restrict__ Wr,
    const float* __restrict__ bias, float* __restrict__ Out,
    int K, int Ndim, int doRelu) {
  // A: 64x16 @ stride 20 (b128-stage aligned, conflict-free ds_load_b64 reads)
  // B: 16x64 @ stride 72 (half-wave K rows land 16 banks apart: disjoint)
  __shared__ float As[2][64 * 20];
  __shared__ float Bs[2][16 * 72];

  const int tid = threadIdx.x;
  const int lane = tid & 31;
  const int wave = tid >> 5;
  const int wm = wave & 3;   // wave row (0..3)
  const int wn = wave >> 2;  // wave col (0..1)
  const int bM = blockIdx.y * 64;
  const int bN = blockIdx.x * 64;
  const int half = lane >> 4;  // 0: lanes 0-15, 1: lanes 16-31
  const int l16 = lane & 15;

  const int SPP = K / KB;  // K-stages per phase
  const int S = 2 * SPP;   // total stages (both A/W sources)

  v8f acc0 = {0.f, 0.f, 0.f, 0.f, 0.f, 0.f, 0.f, 0.f};
  v8f acc1 = {0.f, 0.f, 0.f, 0.f, 0.f, 0.f, 0.f, 0.f};

  // staging assignment: one b128 each for A and B per thread per stage
  const int ra = tid >> 2;        // 0..63
  const int ca = (tid & 3) * 4;   // 0,4,8,12
  int gr = bM + ra;
  if (gr >= N_NODES) gr = N_NODES - 1;  // clamp: masked rows at store anyway
  const int rb = tid >> 4;        // 0..15
  const int cb = (tid & 15) * 4;  // 0..60

  auto issue = [&](int s, int buf) {
    const int ph = (s >= SPP) ? 1 : 0;
    const int k0 = (s - ph * SPP) * KB;
    const float* A = ph ? H : Agg;
    const float* W = ph ? Wr : Wl;
#if USE_ASYNC_LDS
    __builtin_amdgcn_global_load_async_to_lds_b128(
        GPTR(A + (size_t)gr * K + k0 + ca), LPTR(&As[buf][ra * 20 + ca]), 0, 0);
    __builtin_amdgcn_global_load_async_to_lds_b128(
        GPTR(W + (size_t)(k0 + rb) * Ndim + bN + cb),
        LPTR(&Bs[buf][rb * 72 + cb]), 0, 0);
#else
    const float4 va = *(const float4*)(A + (size_t)gr * K + k0 + ca);
    *(float4*)&As[buf][ra * 20 + ca] = va;
    const float4 vb = *(const float4*)(W + (size_t)(k0 + rb) * Ndim + bN + cb);
    *(float4*)&Bs[buf][rb * 72 + cb] = vb;
#endif
  };

  issue(0, 0);
  for (int s = 0; s < S; ++s) {
    const int cur = s & 1;
    const bool more = (s + 1 < S);
    if (more) issue(s + 1, cur ^ 1);  // keep next stage in flight
#if USE_ASYNC_LDS
    if (more) async_wait_le2();  // current stage (2 ops) landed; next still flying
    else     async_wait_le0();
#endif
    __syncthreads();

    const float* as = As[cur];
    const float* bs = Bs[cur];
    const int arow = (wm * 16 + l16) * 20;
    const int bcol = wn * 32 + l16;
#pragma unroll
    for (int kk = 0; kk < KB; kk += 4) {
      const int ka = kk + half * 2;  // lanes 0-15: K=kk,kk+1 ; 16-31: kk+2,kk+3
      v2f a;
      a.x = as[arow + ka];
      a.y = as[arow + ka + 1];
      v2f b0;
      b0.x = bs[ka * 72 + bcol];
      b0.y = bs[(ka + 1) * 72 + bcol];
      acc0 = __builtin_amdgcn_wmma_f32_16x16x4_f32(false, a, false, b0,
                                                   (short)0, acc0, false, false);
      v2f b1;
      b1.x = bs[ka * 72 + bcol + 16];
      b1.y = bs[(ka + 1) * 72 + bcol + 16];
      acc1 = __builtin_amdgcn_wmma_f32_16x16x4_f32(false, a, false, b1,
                                                   (short)0, acc1, false, false);
    }
    __syncthreads();  // all waves done with buf `cur` before it is re-staged
  }

  // ---- epilogue: +bias, optional ReLU, store ----
  const int c0 = bN + wn * 32 + l16;
  const int c1 = c0 + 16;
  const float bb0 = bias[c0];
  const float bb1 = bias[c1];
#pragma unroll
  for (int r = 0; r < 8; ++r) {
    const int row = bM + wm * 16 + half * 8 + r;  // C/D: VGPR r -> M=r / M=8+r
    if (row < N_NODES) {
      float v0 = acc0[r] + bb0;
      float v1 = acc1[r] + bb1;
      if (doRelu) {
        v0 = fmaxf(v0, 0.f);
        v1 = fmaxf(v1, 0.f);
      }
      Out[(size_t)row * Ndim + c0] = v0;
      Out[(size_t)row * Ndim + c1] = v1;
    }
  }
}

// ---------------- BatchNorm (over node axis) ----------------
__global__ void bn_stats_kernel(const float* __restrict__ h,
                                const float* __restrict__ g,
                                const float* __restrict__ beta,
                                float* __restrict__ scale,
                                float* __restrict__ shift, int dout) {
  int j = blockIdx.x * blockDim.x + threadIdx.x;
  if (j >= dout) return;
  float s = 0.f, ss = 0.f;
#pragma unroll 4
  for (int n = 0; n < N_NODES; ++n) {
    float v = h[(size_t)n * dout + j];
    s += v;
    ss += v * v;
  }
  const float inv = 1.0f / (float)N_NODES;
  float mu = s * inv;
  float var = ss * inv - mu * mu;
  float rstd = rsqrtf(var + 1e-5f);
  float sc = g[j] * rstd;
  scale[j] = sc;
  shift[j] = beta[j] - mu * sc;
}

__global__ void bn_apply_kernel(float* __restrict__ h,
                                const float* __restrict__ scale,
                                const float* __restrict__ shift, int dout) {
  long long tid = (long long)blockIdx.x * blockDim.x + threadIdx.x;
  int dout4 = dout >> 2;
  long long total = (long long)N_NODES * dout4;
  if (tid >= total) return;
  int n = (int)(tid / dout4);
  int f = (int)(tid % dout4) * 4;
  float4 v = *(float4*)(h + (size_t)n * dout + f);
  v.x = v.x * scale[f + 0] + shift[f + 0];
  v.y = v.y * scale[f + 1] + shift[f + 1];
  v.z = v.z * scale[f + 2] + shift[f + 2];
  v.w = v.w * scale[f + 3] + shift[f + 3];
  *(float4*)(h + (size_t)n * dout + f) = v;
}

// ---------------------------------------------------------------------------
extern "C" void kernel_launch(void* const* d_in, const int* in_sizes, int n_in,
                              void* d_out, int out_size, void* d_ws,
                              size_t ws_size, hipStream_t stream) {
  (void)in_sizes; (void)n_in; (void)out_size; (void)ws_size;

  const float* x = (const float*)d_in[0];
  const int* ei = (const int*)d_in[1];
  const int* src = ei;            // edge_index[0]
  const int* dst = ei + N_EDGES;  // edge_index[1]

  // params in dict-insertion order: Wl0,Wr0,b0, ..., Wl8,Wr8,b8, g0,beta0..g6,beta6
  const float* Wl[9]; const float* Wr[9]; const float* bp[9];
  for (int li = 0; li < 9; ++li) {
    Wl[li] = (const float*)d_in[2 + 3 * li];
    Wr[li] = (const float*)d_in[3 + 3 * li];
    bp[li] = (const float*)d_in[4 + 3 * li];
  }
  const float* gp[7]; const float* bt[7];
  for (int bi = 0; bi < 7; ++bi) {
    gp[bi] = (const float*)d_in[29 + 2 * bi];
    bt[bi] = (const float*)d_in[30 + 2 * bi];
  }

  // workspace layout (floats)
  float* hA    = (float*)d_ws;                  // N x 1024 max
  float* hB    = hA + (size_t)N_NODES * 1024;   // N x 1024 max
  float* agg   = hB + (size_t)N_NODES * 1024;   // N x 1024 max
  float* rcnt  = agg + (size_t)N_NODES * 1024;  // N
  float* scale = rcnt + N_NODES;                // 1024
  float* shift = scale + 1024;                  // 1024

  static const int dins[9]  = {64, 256, 512, 1024, 1024, 512, 256, 128, 128};
  static const int douts[9] = {256, 512, 1024, 1024, 512, 256, 128, 64, 64};

  // in-degree reciprocals (shared by every layer)
  (void)hipMemsetAsync(rcnt, 0, N_NODES * sizeof(float), stream);
  count_edges_kernel<<<(N_EDGES + 255) / 256, 256, 0, stream>>>(dst, rcnt);
  make_rcnt_kernel<<<(N_NODES + 255) / 256, 256, 0, stream>>>(rcnt);

  const float* h = x;
  float* bufs[2] = {hA, hB};

  for (int li = 0; li < 7; ++li) {
    const int din = dins[li], dout = douts[li];
    float* o = bufs[li & 1];

    (void)hipMemsetAsync(agg, 0, (size_t)N_NODES * din * sizeof(float), stream);
    long long st = (long long)N_EDGES * (din / 4);
    scatter_add_kernel<<<(int)((st + 255) / 256), 256, 0, stream>>>(
        h, src, dst, agg, din / 4);
    long long sr = (long long)N_NODES * (din / 4);
    scale_rows_kernel<<<(int)((sr + 255) / 256), 256, 0, stream>>>(agg, rcnt, din);

    dim3 grid(dout / 64, (N_NODES + 63) / 64);
    sage_gemm_kernel<<<grid, 256, 0, stream>>>(agg, h, Wl[li], Wr[li], bp[li], o,
                                               din, dout, /*relu=*/1);

    bn_stats_kernel<<<(dout + 255) / 256, 256, 0, stream>>>(o, gp[li], bt[li],
                                                            scale, shift, dout);
    long long ba = (long long)N_NODES * (dout / 4);
    bn_apply_kernel<<<(int)((ba + 255) / 256), 256, 0, stream>>>(o, scale, shift,
                                                                 dout);
    h = o;
  }

  // heads: mu (layer 7) and logvar (layer 8) share one aggregation of h
  {
    const int din = 128, dout = 64;
    (void)hipMemsetAsync(agg, 0, (size_t)N_NODES * din * sizeof(float), stream);
    long long st = (long long)N_EDGES * (din / 4);
    scatter_add_kernel<<<(int)((st + 255) / 256), 256, 0, stream>>>(
        h, src, dst, agg, din / 4);
    long long sr = (long long)N_NODES * (din / 4);
    scale_rows_kernel<<<(int)((sr + 255) / 256), 256, 0, stream>>>(agg, rcnt, din);

    float* out_mu = (float*)d_out;
    float* out_lv = out_mu + (size_t)N_NODES * dout;
    dim3 grid(dout / 64, (N_NODES + 63) / 64);
    sage_gemm_kernel<<<grid, 256, 0, stream>>>(agg, h, Wl[7], Wr[7], bp[7],
                                               out_mu, din, dout, /*relu=*/0);
    sage_gemm_kernel<<<grid, 256, 0, stream>>>(agg, h, Wl[8], Wr[8], bp[8],
                                               out_lv, din, dout, /*relu=*/0);
  }
}